// CausalSelfAttention_81295140978734
// MI455X (gfx1250) — compile-verified
//
#include <hip/hip_runtime.h>

// MI455X / gfx1250, wave32. All matrix math on V_WMMA_F32_16X16X4_F32 (fp32
// reference => fp32 WMMA keeps numerics while using the matrix pipes).
// Fragment pairs (even/odd K) are kept adjacent in LDS/global so every
// operand is a single aligned b64 load into the even-aligned VGPR pair.

typedef __attribute__((ext_vector_type(2))) float v2f;
typedef __attribute__((ext_vector_type(8))) float v8f;

static constexpr int Tt = 2048;   // sequence length
static constexpr int Cc = 1024;   // channels
static constexpr int Hh = 16;     // heads
static constexpr int HD = 64;     // head dim
static constexpr int BLOCK = 2048;

static __device__ inline v8f wmma4(v2f a, v2f b, v8f c) {
  // D = A(16x4) * B(4x16) + C(16x16), fp32
  return __builtin_amdgcn_wmma_f32_16x16x4_f32(
      /*neg_a=*/false, a, /*neg_b=*/false, b,
      /*c_mod=*/(short)0, c, /*reuse_a=*/false, /*reuse_b=*/false);
}

static __device__ inline v8f vzero8() {
  v8f v;
#pragma unroll
  for (int i = 0; i < 8; ++i) v[i] = 0.0f;
  return v;
}

// ---------------------------------------------------------------- prep ------

__global__ void zero_f32(float* p, int n) {
  int i = blockIdx.x * blockDim.x + threadIdx.x;
  if (i < n) p[i] = 0.0f;
}

__global__ void hist_kernel(const int* __restrict__ tok, float* __restrict__ cnt) {
  int t = blockIdx.x * blockDim.x + threadIdx.x;
  if (t < Tt) atomicAdd(&cnt[tok[t]], 1.0f);
}

// single workgroup: per_tok = 1/(cnt[idx]+eps); partial = inclusive cumsum;
// also gather padding mask pmg[t] = padding_mask[token_index[t]]
__global__ __launch_bounds__(256) void scan_kernel(
    const float* __restrict__ cnt, const int* __restrict__ tok,
    const float* __restrict__ pmask, float* __restrict__ partial,
    float* __restrict__ pmg) {
  __shared__ float sums[256];
  __shared__ float off[256];
  int tid = threadIdx.x;
  float local[8];
  float s = 0.0f;
  int base = tid * 8;
#pragma unroll
  for (int i = 0; i < 8; ++i) {
    int t = base + i;
    float pt = 1.0f / (cnt[tok[t]] + 1e-10f);
    s += pt;
    local[i] = s;  // inclusive within chunk
  }
  sums[tid] = s;
  __syncthreads();
  if (tid == 0) {
    float a = 0.0f;
    for (int i = 0; i < 256; ++i) { off[i] = a; a += sums[i]; }
  }
  __syncthreads();
  float o = off[tid];
#pragma unroll
  for (int i = 0; i < 8; ++i) partial[base + i] = local[i] + o;
  for (int t = tid; t < Tt; t += 256) pmg[t] = pmask[tok[t]];
}

__global__ void tables_kernel(const float* __restrict__ partial,
                              float* __restrict__ cost, float* __restrict__ sint) {
  int idx = blockIdx.x * blockDim.x + threadIdx.x;  // Tt*32 = 65536
  int t = idx >> 5, i = idx & 31;
  // inv_freq[i] = 10000^(-i/32)
  float invf = __expf(-(float)i * (9.210340371976184f / 32.0f));
  float f = partial[t] * invf;
  cost[idx] = __cosf(f);
  sint[idx] = __sinf(f);
}

// RoPE + overrides + v scaling.
// q,k written as [H][T][HD] (row-major per head).
// v written PAIR-INTERLEAVED: vh[(((h*(T/2) + t/2)*HD) + d)*2 + (t&1)]
// so the P.V B-fragment (even/odd key pair, same dim) is one contiguous v2f.
__global__ __launch_bounds__(256) void rope_kernel(
    const float* __restrict__ qkv, const float* __restrict__ cost,
    const float* __restrict__ sint, const float* __restrict__ cs,
    float* __restrict__ qh, float* __restrict__ kh, float* __restrict__ vh) {
  int idx = blockIdx.x * blockDim.x + threadIdx.x;  // Tt*Hh = 32768
  int t = idx >> 4, h = idx & 15;
  const float* row = qkv + (size_t)t * (3 * Cc);
  float c_s = cs[t];
  float ev = __expf(c_s);
  size_t ob = ((size_t)h * Tt + t) * HD;
  size_t vb0 = (((size_t)h * (Tt / 2) + (t >> 1)) * HD) * 2 + (t & 1);
#pragma unroll 4
  for (int d = 0; d < 32; ++d) {
    float co = cost[t * 32 + d], si = sint[t * 32 + d];
    float qa = row[h * HD + d],          qb = row[h * HD + d + 32];
    float ka = row[Cc + h * HD + d],     kb = row[Cc + h * HD + d + 32];
    float va = row[2 * Cc + h * HD + d], vb = row[2 * Cc + h * HD + d + 32];
    float qlo = qa * co - qb * si;
    float qhi = qb * co + qa * si;
    float klo = ka * co - kb * si;
    float khi = kb * co + ka * si;
    if (d == 31) { qhi = 1.0f; khi = c_s; }  // q[...,-1]=1, k[...,-1]=cum_scores
    qh[ob + d] = qlo; qh[ob + d + 32] = qhi;
    kh[ob + d] = klo; kh[ob + d + 32] = khi;
    vh[vb0 + (size_t)d * 2] = va * ev;
    vh[vb0 + (size_t)(d + 32) * 2] = vb * ev;
  }
}

// ---------------------------------------------------------------- GEMM ------
// C[M,N] = A[M,K] @ B[K,N] + bias[N]; M,N multiples of 64, K multiple of 32.
// 4 waves per WG; WG tile 64x64; wave computes a 16x64 strip (4 accum tiles).
// B is staged pair-interleaved: BsP[kp][n][2]; row stride 160 floats puts the
// hi-half lanes (kp+1) on banks 32-63 vs 0-31 -> conflict-free ds_load_b64.
static constexpr int BSTRIDE = 160;  // floats per kp row (128 data + 32 pad)

__global__ __launch_bounds__(128) void gemm_f32_wmma(
    const float* __restrict__ A, const float* __restrict__ B,
    const float* __restrict__ bias, float* __restrict__ C,
    int M, int N, int K) {
  __shared__ float As[64][36];            // +4 pad: conflict-free column reads
  __shared__ float BsP[16 * BSTRIDE];     // 16 kp rows of 64 interleaved pairs
  const int tid = threadIdx.x;
  const int lane = tid & 31, wave = tid >> 5;
  const bool hi = lane >= 16;
  const int l15 = lane & 15;
  const int m0 = blockIdx.y * 64, n0 = blockIdx.x * 64;

  v8f acc[4];
#pragma unroll
  for (int t = 0; t < 4; ++t) acc[t] = vzero8();

  for (int k0 = 0; k0 < K; k0 += 32) {
    // stage A 64x32 (float4, coalesced)
    for (int i = tid; i < 512; i += 128) {
      int r = i >> 3, c4 = (i & 7) << 2;
      const float* src = &A[(size_t)(m0 + r) * K + k0 + c4];
      float4 v = *(const float4*)src;
      *(float4*)&As[r][c4] = v;
      if (k0 + 32 < K) __builtin_prefetch(src + 32, 0, 0);
    }
    // stage B 32x64 -> pair-interleaved BsP[kp][n][r&1]
    for (int i = tid; i < 512; i += 128) {
      int r = i >> 4, c4 = (i & 15) << 2;
      const float* src = &B[(size_t)(k0 + r) * N + n0 + c4];
      float4 v = *(const float4*)src;
      float* dst = &BsP[(r >> 1) * BSTRIDE + (c4 << 1) + (r & 1)];
      dst[0] = v.x; dst[2] = v.y; dst[4] = v.z; dst[6] = v.w;
      if (k0 + 32 < K) __builtin_prefetch(src + (size_t)32 * N, 0, 0);
    }
    __syncthreads();
#pragma unroll
    for (int kk = 0; kk < 32; kk += 4) {
      // A frag: lanes0-15 rows M, K=kk+0/1; lanes16-31 K=kk+2/3 (one b64)
      v2f a = *(const v2f*)&As[wave * 16 + l15][kk + (hi ? 2 : 0)];
      const int kp = (kk >> 1) + (hi ? 1 : 0);
#pragma unroll
      for (int t = 0; t < 4; ++t) {
        // B frag: lane n col N, even/odd K pair adjacent (one b64)
        v2f b = *(const v2f*)&BsP[kp * BSTRIDE + ((t * 16 + l15) << 1)];
        acc[t] = wmma4(a, b, acc[t]);
      }
    }
    __syncthreads();
  }
#pragma unroll
  for (int t = 0; t < 4; ++t) {
    int col = n0 + t * 16 + l15;
    float bv = bias ? bias[col] : 0.0f;
#pragma unroll
    for (int r = 0; r < 8; ++r) {  // C layout: vgpr r -> row r (lanes<16) / r+8
      int row = m0 + wave * 16 + r + (hi ? 8 : 0);
      C[(size_t)row * N + col] = acc[t][r] + bv;
    }
  }
}

// ------------------------------------------------------------- attention ----
// One WG = one (head, 16-row Q block). Full 16x2048 score strip lives in
// 128 KB LDS (CDNA5: 320 KB/WGP) -> plain softmax, no online rescaling.
__global__ __launch_bounds__(256) void attn_kernel(
    const float* __restrict__ qh, const float* __restrict__ kh,
    const float* __restrict__ vh, const float* __restrict__ pmg,
    float* __restrict__ yout) {
  __shared__ float sc[16][Tt];      // 128 KB score/prob strip
  __shared__ float red[16][17];
  __shared__ float rowm[16];
  __shared__ float rowsum[16];
  __shared__ float ypart[4][16][16];

  const int qb = blockIdx.x;        // 0..127
  const int h = blockIdx.y;         // 0..15
  const int q0 = qb * 16;
  const int tid = threadIdx.x, lane = tid & 31, wave = tid >> 5;
  const bool hi = lane >= 16;
  const int l15 = lane & 15;
  const float* Q = qh + (size_t)h * Tt * HD;
  const float* Kp = kh + (size_t)h * Tt * HD;
  const float* Vp = vh + (size_t)h * Tt * HD;   // pair-interleaved layout
  const float NEG = -__builtin_inff();

  // preload Q A-fragments for all 16 K-steps (HD=64, K-step 4)
  v2f qa[16];
  {
    const float* qrow = Q + (size_t)(q0 + l15) * HD;
#pragma unroll
    for (int s = 0; s < 16; ++s)
      qa[s] = *(const v2f*)&qrow[4 * s + (hi ? 2 : 0)];
  }

  // ---- phase 1: S = Q K^T / 8 + mask, into LDS ----
  for (int j = wave; j < Tt / 16; j += 8) {  // wave-uniform tiles
    int kbase = j * 16;
    if (kbase > q0 + 15) {  // fully above diagonal
#pragma unroll
      for (int r = 0; r < 8; ++r) sc[r + (hi ? 8 : 0)][kbase + l15] = NEG;
      continue;
    }
    v8f acc = vzero8();
    const float* krow = Kp + (size_t)(kbase + l15) * HD;
#pragma unroll
    for (int s = 0; s < 16; ++s) {
      v2f b = *(const v2f*)&krow[4 * s + (hi ? 2 : 0)];
      acc = wmma4(qa[s], b, acc);
    }
    float pk = pmg[kbase + l15];
#pragma unroll
    for (int r = 0; r < 8; ++r) {
      int rM = r + (hi ? 8 : 0);
      int qi = q0 + rM;
      int ki = kbase + l15;
      float v = acc[r] * 0.125f;  // 1/sqrt(64)
      if (ki > qi || (pmg[qi] * pk) == 0.0f) v = NEG;
      sc[rM][ki] = v;
    }
  }
  __syncthreads();

  // ---- phase 2: row softmax in LDS (16 threads per row) ----
  {
    int row = tid >> 4, sub = tid & 15;
    float m = NEG;
    for (int c = sub; c < Tt; c += 16) m = fmaxf(m, sc[row][c]);
    red[row][sub] = m;
    __syncthreads();
    if (sub == 0) {
      float mm = red[row][0];
      for (int i = 1; i < 16; ++i) mm = fmaxf(mm, red[row][i]);
      rowm[row] = mm;
    }
    __syncthreads();
    float rm = rowm[row];
    float s = 0.0f;
    for (int c = sub; c < Tt; c += 16) {
      float e = __expf(sc[row][c] - rm);
      sc[row][c] = e;
      s += e;
    }
    red[row][sub] = s;
    __syncthreads();
    if (sub == 0) {
      float ss = 0.0f;
      for (int i = 0; i < 16; ++i) ss += red[row][i];
      rowsum[row] = ss;
    }
    __syncthreads();
    float inv = 1.0f / rowsum[row];
    for (int c = sub; c < Tt; c += 16) sc[row][c] *= inv;
  }
  __syncthreads();

  // ---- phase 3: Y = P V (only tiles up to the diagonal carry weight) ----
  const int nt = wave & 3;    // which 16-wide slice of HD
  const int half = wave >> 2; // split key range across wave pairs
  v8f acc = vzero8();
  const int jmax = qb;
  for (int j = half; j <= jmax; j += 2) {
#pragma unroll
    for (int s = 0; s < 4; ++s) {  // K=16 keys per tile -> 4 WMMA steps
      // P frag from LDS: even/odd key pair contiguous (one b64)
      v2f a = *(const v2f*)&sc[l15][j * 16 + 4 * s + (hi ? 2 : 0)];
      // V frag: pair-interleaved layout -> one coalesced global b64
      int kp = (j * 16 + 4 * s + (hi ? 2 : 0)) >> 1;
      v2f b = *(const v2f*)&Vp[(((size_t)kp * HD) + nt * 16 + l15) * 2];
      acc = wmma4(a, b, acc);
    }
  }
  if (half == 1) {
#pragma unroll
    for (int r = 0; r < 8; ++r) ypart[nt][r + (hi ? 8 : 0)][l15] = acc[r];
  }
  __syncthreads();
  if (half == 0) {
#pragma unroll
    for (int r = 0; r < 8; ++r) {
      int rM = r + (hi ? 8 : 0);
      float v = acc[r] + ypart[nt][rM][l15];
      yout[(size_t)(q0 + rM) * Cc + h * HD + nt * 16 + l15] = v;
    }
  }
}

// ---------------------------------------------------------------- launch ----

extern "C" void kernel_launch(void* const* d_in, const int* in_sizes, int n_in,
                              void* d_out, int out_size, void* d_ws, size_t ws_size,
                              hipStream_t stream) {
  const float* x      = (const float*)d_in[0];
  const float* cs     = (const float*)d_in[1];
  const float* pmask  = (const float*)d_in[2];
  const int*   tok    = (const int*)d_in[3];
  const float* w_attn = (const float*)d_in[4];
  const float* b_attn = (const float*)d_in[5];
  const float* w_proj = (const float*)d_in[6];
  const float* b_proj = (const float*)d_in[7];
  float* out = (float*)d_out;

  // workspace layout (floats): ~59 MB total
  float* ws      = (float*)d_ws;
  float* qkv     = ws;                          // 2048*3072
  float* qh      = qkv + (size_t)Tt * 3 * Cc;   // 16*2048*64 each
  float* kh      = qh + (size_t)Hh * Tt * HD;
  float* vh      = kh + (size_t)Hh * Tt * HD;   // pair-interleaved
  float* yout    = vh + (size_t)Hh * Tt * HD;   // 2048*1024
  float* cost    = yout + (size_t)Tt * Cc;      // 2048*32
  float* sint    = cost + Tt * 32;
  float* cnt     = sint + Tt * 32;              // 2048
  float* partial = cnt + BLOCK;                 // 2048
  float* pmg     = partial + Tt;                // 2048

  zero_f32<<<(BLOCK + 255) / 256, 256, 0, stream>>>(cnt, BLOCK);
  hist_kernel<<<Tt / 256, 256, 0, stream>>>(tok, cnt);
  scan_kernel<<<1, 256, 0, stream>>>(cnt, tok, pmask, partial, pmg);
  tables_kernel<<<(Tt * 32) / 256, 256, 0, stream>>>(partial, cost, sint);

  // qkv = x @ w_attn + b_attn   (2048 x 3072, K=1024)
  gemm_f32_wmma<<<dim3(3 * Cc / 64, Tt / 64), 128, 0, stream>>>(
      x, w_attn, b_attn, qkv, Tt, 3 * Cc, Cc);

  rope_kernel<<<(Tt * Hh) / 256, 256, 0, stream>>>(qkv, cost, sint, cs, qh, kh, vh);

  attn_kernel<<<dim3(Tt / 16, Hh), 256, 0, stream>>>(qh, kh, vh, pmg, yout);

  // out = yout @ w_proj + b_proj   (2048 x 1024, K=1024)
  gemm_f32_wmma<<<dim3(Cc / 64, Tt / 64), 128, 0, stream>>>(
      yout, w_proj, b_proj, out, Tt, Cc, Cc);
}